// Heston_70901320122437
// MI455X (gfx1250) — compile-verified
//
#include <hip/hip_runtime.h>
#include <hip/hip_bf16.h>
#include <cstdint>
#include <cstddef>

typedef __attribute__((ext_vector_type(2))) float v2f;
typedef __attribute__((ext_vector_type(8))) float v8f;
// Exact pointee type of the async-LDS builtins per hipcc diagnostic:
// "__attribute__((__vector_size__(4 * sizeof(int)))) int"
typedef int v4i_vs __attribute__((__vector_size__(4 * sizeof(int))));

#define NSTEPS  360
#define NSTRIKE 46

#define AS1 __attribute__((address_space(1)))
#define AS3 __attribute__((address_space(3)))

#if __has_builtin(__builtin_amdgcn_global_load_async_to_lds_b128) && \
    __has_builtin(__builtin_amdgcn_s_wait_asynccnt)
#define HAVE_ASYNC 1
#else
#define HAVE_ASYNC 0
#endif

// ---------------------------------------------------------------------------
// Shared per-step Heston update (order matches reference: sqV from old V,
// S_new from old S/V, V_new from old V, then clamp).
// ---------------------------------------------------------------------------
__device__ __forceinline__ void heston_step(float& S, float& V,
                                            float zj, float z1j,
                                            float rh, float sqh, float h) {
  const float sqV = __builtin_amdgcn_sqrtf(V);
  float Sn = fmaf(S, rh, S);                 // S + S*r*h
  Sn = fmaf(S * sqV, sqh * z1j, Sn);         // + S*sqV*(sqh*z1)
  float Vn = fmaf(0.04f - V, 1.5f * h, V);   // V + 1.5*(0.04-V)*h
  Vn = fmaf(0.3f * sqV, sqh * zj, Vn);       // + 0.3*sqV*(sqh*z)
  V = fmaxf(Vn, 0.0f);
  S = Sn;
}

// ---------------------------------------------------------------------------
// Kernel 1a (fallback): one thread per path, float4 row streaming + prefetch.
// ---------------------------------------------------------------------------
__global__ void heston_sim(const float* __restrict__ S0,
                           const float* __restrict__ V0,
                           const float* __restrict__ rate,
                           const float* __restrict__ z,
                           const float* __restrict__ z1,
                           const int*   __restrict__ idx,
                           float*       __restrict__ snap,
                           int MC, int nsnap) {
  int path = blockIdx.x * blockDim.x + threadIdx.x;
  if (path >= MC) return;

  const float h   = 1.0f / (float)NSTEPS;
  const float sqh = __builtin_amdgcn_sqrtf(h);
  const float r   = rate[0];
  const float rh  = r * h;

  float S = S0[0];
  float V = V0[0];

  const float* zr  = z  + (size_t)path * NSTEPS;
  const float* z1r = z1 + (size_t)path * NSTEPS;

  int next     = 0;
  int nextStep = (nsnap > 0) ? idx[0] : 0x7fffffff;

  for (int t = 0; t < NSTEPS; t += 4) {
    if (t + 16 < NSTEPS) {
      __builtin_prefetch(zr  + t + 16, 0, 3);
      __builtin_prefetch(z1r + t + 16, 0, 3);
    }
    const float4 zv  = *(const float4*)(zr  + t);
    const float4 z1v = *(const float4*)(z1r + t);
#pragma unroll
    for (int j = 0; j < 4; ++j) {
      const float zj  = (j == 0) ? zv.x  : (j == 1) ? zv.y  : (j == 2) ? zv.z  : zv.w;
      const float z1j = (j == 0) ? z1v.x : (j == 1) ? z1v.y : (j == 2) ? z1v.z : z1v.w;
      heston_step(S, V, zj, z1j, rh, sqh, h);
      if (t + j + 1 == nextStep) {            // uniform branch
        snap[(size_t)next * MC + path] = S;
        ++next;
        nextStep = (next < nsnap) ? idx[next] : 0x7fffffff;
      }
    }
  }
}

#if HAVE_ASYNC
// ---------------------------------------------------------------------------
// Kernel 1b: async-to-LDS tiled streamer. One wave per block; the wave owns 32
// consecutive paths. z/z1 are row-major per path (stride 1440 B), so per-lane
// streaming scatters 32 lines per issue. Instead, the wave cooperatively
// fetches a 32-path x 36-step tile with GLOBAL_LOAD_ASYNC_TO_LDS_B128 using a
// flattened (path,chunk) -> lane map: 9 consecutive lanes cover 144 B of one
// path row (line-contiguous), double-buffered on ASYNCcnt. 360 = 10*36 exactly.
// LDS rows padded to 40 dwords (160 B) -> 16 B-aligned B128 ds_loads, <=2-way
// bank conflicts. 20.5 KB LDS per wave-block.
// ---------------------------------------------------------------------------
#define TS   36          // steps per tile
#define NCHK 9           // B128 chunks per path row per tile (TS/4)
#define RS   40          // padded LDS row stride in dwords

__device__ __forceinline__ void async_b128(const float* g, float* l) {
  __builtin_amdgcn_global_load_async_to_lds_b128(
      (AS1 v4i_vs*)g, (AS3 v4i_vs*)l, 0, 0);
}

__global__ void __launch_bounds__(32)
heston_sim_async(const float* __restrict__ S0,
                 const float* __restrict__ V0,
                 const float* __restrict__ rate,
                 const float* __restrict__ z,
                 const float* __restrict__ z1,
                 const int*   __restrict__ idx,
                 float*       __restrict__ snap,
                 int MC, int nsnap) {
  __shared__ __align__(16) float tz [2][32 * RS];
  __shared__ __align__(16) float tz1[2][32 * RS];

  const int lane     = threadIdx.x;            // 0..31
  const int pathBase = blockIdx.x * 32;
  const int path     = pathBase + lane;

  // Per-lane transfer slots: T = i*32+lane, p = T/NCHK, c = T%NCHK.
  int goff[NCHK], loff[NCHK];
#pragma unroll
  for (int i = 0; i < NCHK; ++i) {
    const int T = i * 32 + lane;
    const int p = T / NCHK, c = T % NCHK;
    goff[i] = p * NSTEPS + c * 4;              // dword offset in global block
    loff[i] = p * RS + c * 4;                  // dword offset in LDS tile
  }

  const float* zb  = z  + (size_t)pathBase * NSTEPS;
  const float* z1b = z1 + (size_t)pathBase * NSTEPS;

  // Prime the pipeline: tile 0 -> buffer 0 (2*NCHK async ops).
#pragma unroll
  for (int i = 0; i < NCHK; ++i) {
    async_b128(zb  + goff[i], &tz[0][loff[i]]);
    async_b128(z1b + goff[i], &tz1[0][loff[i]]);
  }

  const float h   = 1.0f / (float)NSTEPS;
  const float sqh = __builtin_amdgcn_sqrtf(h);
  const float r   = rate[0];
  const float rh  = r * h;
  float S = S0[0];
  float V = V0[0];

  int next     = 0;
  int nextStep = (nsnap > 0) ? idx[0] : 0x7fffffff;

  const int NT = NSTEPS / TS;                  // 10 tiles, no tail
  for (int tile = 0; tile < NT; ++tile) {
    const int buf = tile & 1;
    if (tile + 1 < NT) {
      const int nb  = buf ^ 1;
      const int t0n = (tile + 1) * TS;
#pragma unroll
      for (int i = 0; i < NCHK; ++i) {
        async_b128(zb  + t0n + goff[i], &tz[nb][loff[i]]);
        async_b128(z1b + t0n + goff[i], &tz1[nb][loff[i]]);
      }
      asm volatile("" ::: "memory");
      __builtin_amdgcn_s_wait_asynccnt(2 * NCHK);   // current tile complete
    } else {
      asm volatile("" ::: "memory");
      __builtin_amdgcn_s_wait_asynccnt(0);          // last tile complete
    }
    asm volatile("" ::: "memory");

    const float* lz = &tz[buf][lane * RS];
    const float* l1 = &tz1[buf][lane * RS];
#pragma unroll
    for (int s = 0; s < TS; s += 4) {
      const float4 zv  = *(const float4*)(lz + s);   // ds_load_b128
      const float4 z1v = *(const float4*)(l1 + s);
#pragma unroll
      for (int j = 0; j < 4; ++j) {
        const float zj  = (j == 0) ? zv.x  : (j == 1) ? zv.y  : (j == 2) ? zv.z  : zv.w;
        const float z1j = (j == 0) ? z1v.x : (j == 1) ? z1v.y : (j == 2) ? z1v.z : z1v.w;
        heston_step(S, V, zj, z1j, rh, sqh, h);
        const int t = tile * TS + s + j;
        if (t + 1 == nextStep) {                     // uniform branch
          snap[(size_t)next * MC + path] = S;
          ++next;
          nextStep = (next < nsnap) ? idx[next] : 0x7fffffff;
        }
      }
    }
  }
}
#endif  // HAVE_ASYNC

// ---------------------------------------------------------------------------
// Kernel 2: WMMA payoff reduction (lowered to 12 in-loop
// v_wmma_f32_16x16x4_f32 in round 1). One wave32 per block, EXEC all-ones.
// ---------------------------------------------------------------------------
__global__ void payoff_wmma(const float* __restrict__ snap,
                            float*       __restrict__ partial,
                            int MC, int nchunk) {
  const int lane  = threadIdx.x;     // 0..31
  const int t     = blockIdx.y;
  const int chunk = blockIdx.x;
  const int per   = MC / nchunk;     // multiple of 4 (host guarantees)
  const int base0 = chunk * per;
  const int m     = lane & 15;
  const int half  = lane >> 4;

  const float* St = snap + (size_t)t * MC;

  float K[12];
#pragma unroll
  for (int sb = 0; sb < 3; ++sb) {
    const int  row = sb * 16 + m;
    const bool pad = (row >= NSTRIKE);
    const float kc = 100.0f + (float)row;
    const float kp =  55.0f + (float)row;
    K[0 * 3 + sb] = pad ?  1e30f : kc;   // call @ Kc
    K[1 * 3 + sb] = pad ? -1e30f : kp;   // put  @ Kp
    K[2 * 3 + sb] = pad ?  1e30f : kp;   // call @ Kp
    K[3 * 3 + sb] = pad ? -1e30f : kc;   // put  @ Kc
  }

  const v8f vzero = {};
  v8f acc[12];
#pragma unroll
  for (int g = 0; g < 12; ++g) acc[g] = vzero;

  const v2f Bones = {1.0f, 1.0f};

  for (int i = 0; i < per; i += 4) {
    const float4 sv = *(const float4*)(St + base0 + i);
    const float s0 = half ? sv.z : sv.x;
    const float s1 = half ? sv.w : sv.y;
#pragma unroll
    for (int g = 0; g < 12; ++g) {
      const int  kind   = g / 3;
      const bool isCall = (kind == 0) || (kind == 2);
      const float k = K[g];
      v2f A;
      if (isCall) { A.x = fmaxf(s0 - k, 0.0f); A.y = fmaxf(s1 - k, 0.0f); }
      else        { A.x = fmaxf(k - s0, 0.0f); A.y = fmaxf(k - s1, 0.0f); }
      acc[g] = __builtin_amdgcn_wmma_f32_16x16x4_f32(
          false, A, false, Bones, (short)0, acc[g], false, false);
    }
  }

  if (m == 0) {
    const size_t pb = (((size_t)t * nchunk + chunk) * 12) * 16;
#pragma unroll
    for (int g = 0; g < 12; ++g) {
#pragma unroll
      for (int v = 0; v < 8; ++v) {
        partial[pb + (size_t)g * 16 + half * 8 + v] = acc[g][v];
      }
    }
  }
}

// ---------------------------------------------------------------------------
// Kernel 3: deterministic finalize. out[kind][t][k] = (sum partials)/MC * disc
// ---------------------------------------------------------------------------
__global__ void finalize_k(const float* __restrict__ partial,
                           const float* __restrict__ rate,
                           const int*   __restrict__ idx,
                           float*       __restrict__ out,
                           int MC, int nchunk, int nsnap) {
  const int tid   = blockIdx.x * blockDim.x + threadIdx.x;
  const int total = 4 * nsnap * NSTRIKE;
  if (tid >= total) return;
  const int k    = tid % NSTRIKE;
  const int t    = (tid / NSTRIKE) % nsnap;
  const int kind = tid / (NSTRIKE * nsnap);
  const int sb   = k >> 4;
  const int mrow = k & 15;
  const int g    = kind * 3 + sb;

  float sum = 0.0f;
  for (int c = 0; c < nchunk; ++c) {
    sum += partial[(((size_t)t * nchunk + c) * 12 + g) * 16 + mrow];
  }
  const float r    = rate[0];
  const float disc = expf(-r * (float)idx[t] / (float)NSTEPS);
  out[tid] = sum / (float)MC * disc;
}

// ---------------------------------------------------------------------------
extern "C" void kernel_launch(void* const* d_in, const int* in_sizes, int n_in,
                              void* d_out, int out_size, void* d_ws, size_t ws_size,
                              hipStream_t stream) {
  const float* S0   = (const float*)d_in[0];
  const float* V0   = (const float*)d_in[1];
  const float* rate = (const float*)d_in[2];
  const float* z    = (const float*)d_in[3];
  const float* z1   = (const float*)d_in[4];
  const int*   idx  = (const int*)d_in[5];

  const int MC    = in_sizes[3] / NSTEPS;   // z is (MC, NSTEPS)
  const int nsnap = in_sizes[5];
  if (MC <= 0 || nsnap <= 0) return;

  float* out  = (float*)d_out;
  float* snap = (float*)d_ws;
  const size_t snapBytes = (size_t)nsnap * (size_t)MC * sizeof(float);

  int nchunk = 128;
  while (nchunk > 1 && (MC % (nchunk * 4) != 0)) nchunk >>= 1;
  while (nchunk > 1 &&
         snapBytes + (size_t)nsnap * nchunk * 12 * 16 * sizeof(float) > ws_size)
    nchunk >>= 1;

  float* partial = (float*)((char*)d_ws + snapBytes);

#if HAVE_ASYNC
  if (MC % 32 == 0) {
    heston_sim_async<<<MC / 32, 32, 0, stream>>>(S0, V0, rate, z, z1, idx,
                                                 snap, MC, nsnap);
  } else
#endif
  {
    heston_sim<<<(MC + 255) / 256, 256, 0, stream>>>(S0, V0, rate, z, z1, idx,
                                                     snap, MC, nsnap);
  }

  dim3 g2(nchunk, nsnap, 1);
  payoff_wmma<<<g2, 32, 0, stream>>>(snap, partial, MC, nchunk);

  const int total = 4 * nsnap * NSTRIKE;
  finalize_k<<<(total + 255) / 256, 256, 0, stream>>>(partial, rate, idx, out,
                                                      MC, nchunk, nsnap);
}